// ReverseLossLayer_82420422410466
// MI455X (gfx1250) — compile-verified
//
#include <hip/hip_runtime.h>
#include <hip/hip_bf16.h>

typedef __attribute__((ext_vector_type(2))) float v2f;
typedef __attribute__((ext_vector_type(4))) float v4f;
typedef __attribute__((ext_vector_type(8))) float v8f;

#define N_SRC 16384
#define M_TAR 16384
#define CHUNK 2048                 // source points staged in LDS per pass (32 KB)
#define THREADS 128                // 4 waves (wave32)
#define WAVES (THREADS / 32)
#define ROWS_PER_BLOCK (WAVES * 16)
#define NBLOCKS (M_TAR / ROWS_PER_BLOCK)   // 256

// ---------------------------------------------------------------------------
// Kernel 1: pack each source point as (x, y, z, ||s||^2) for aligned b128 I/O.
// ---------------------------------------------------------------------------
__global__ void pack_src_kernel(const float* __restrict__ src,
                                v4f* __restrict__ packed, int n) {
    int i = blockIdx.x * blockDim.x + threadIdx.x;
    if (i < n) {
        float x = src[3 * i + 0];
        float y = src[3 * i + 1];
        float z = src[3 * i + 2];
        v4f p;
        p.x = x; p.y = y; p.z = z;
        p.w = x * x + y * y + z * z;
        packed[i] = p;
    }
}

// ---------------------------------------------------------------------------
// Kernel 2: per-target-row argmin of (||s||^2 - 2 t.s) via f32 WMMA 16x16x4.
//   A (16x4)  = [-2*t0, -2*t1, -2*t2, 1]      (held in regs per wave)
//   B (4x16)  = [s0, s1, s2, ||s||^2]         (streamed from LDS)
//   D = A*B + 0  ->  score[m][n]; running min + index tracked per C-slot.
// ---------------------------------------------------------------------------
__global__ __launch_bounds__(THREADS)
void nn_argmin_kernel(const float* __restrict__ src,
                      const float* __restrict__ tar,
                      const v4f* __restrict__ packed,
                      float* __restrict__ blockPartial) {
    __shared__ float ldsPts[CHUNK * 4];     // (x,y,z,norm) per staged point
    __shared__ float wavePartial[WAVES];

    const int tid  = threadIdx.x;
    const int lane = tid & 31;
    const int wave = tid >> 5;
    const int half = lane >> 4;             // 0: K={0,1}, 1: K={2,3}
    const int l16  = lane & 15;

    const int rowTile = blockIdx.x * WAVES + wave;
    const int rowBase = rowTile * 16;

    // ---- A fragment: lane holds row M = rowBase + l16, K = 2*half + {0,1}
    v2f a;
    {
        const float* t = tar + (rowBase + l16) * 3;
        if (half == 0) { a.x = -2.0f * t[0]; a.y = -2.0f * t[1]; }
        else           { a.x = -2.0f * t[2]; a.y = 1.0f;         }
    }

    float bv[8];
    int   bi[8];
#pragma unroll
    for (int i = 0; i < 8; ++i) { bv[i] = 3.0e38f; bi[i] = 0; }

    for (int base = 0; base < N_SRC; base += CHUNK) {
        __syncthreads();                    // previous chunk fully consumed
        for (int p = tid; p < CHUNK; p += THREADS) {
            v4f v = packed[base + p];       // global_load_b128 (L2-resident)
            *(v4f*)(&ldsPts[p << 2]) = v;   // ds_store_b128
        }
        __syncthreads();

#pragma unroll 4
        for (int j = 0; j < CHUNK / 16; ++j) {
            // B fragment: lane holds col N = l16, K = 2*half + {0,1}
            int p  = (j << 4) + l16;
            v2f b  = *(const v2f*)(&ldsPts[(p << 2) + (half << 1)]); // ds_load_b64
            v8f c  = {};                    // C = 0: D = ||s||^2 - 2 t.s
            c = __builtin_amdgcn_wmma_f32_16x16x4_f32(
                    false, a, false, b, (short)0, c, false, false);
            int jj = (base >> 4) + j;       // global column-tile index
#pragma unroll
            for (int i = 0; i < 8; ++i) {   // strict '<' keeps first minimum
                bool lt = c[i] < bv[i];
                bv[i] = lt ? c[i] : bv[i];
                bi[i] = lt ? jj   : bi[i];
            }
        }
    }

    // Recover absolute column per slot; reduce across the 16-lane half.
    int bc[8];
#pragma unroll
    for (int i = 0; i < 8; ++i) bc[i] = (bi[i] << 4) + l16;

#pragma unroll
    for (int m = 1; m <= 8; m <<= 1) {      // stays within each 16-lane half
#pragma unroll
        for (int i = 0; i < 8; ++i) {
            float ov = __shfl_xor(bv[i], m, 32);
            int   oc = __shfl_xor(bc[i], m, 32);
            bool take = (ov < bv[i]) || (ov == bv[i] && oc < bc[i]);
            bv[i] = take ? ov : bv[i];
            bc[i] = take ? oc : bc[i];
        }
    }

    // Lanes 0 / 16 own rows [rowBase..+7] / [rowBase+8..+15]; recompute exact
    // squared distances from the winning indices (matches reference rounding).
    float sum = 0.0f;
    if (l16 == 0) {
#pragma unroll
        for (int i = 0; i < 8; ++i) {
            int row = rowBase + (half << 3) + i;
            int col = bc[i];
            float dx = src[3 * col + 0] - tar[3 * row + 0];
            float dy = src[3 * col + 1] - tar[3 * row + 1];
            float dz = src[3 * col + 2] - tar[3 * row + 2];
            sum += dx * dx + dy * dy + dz * dz;
        }
    }
    // Fixed-order wave reduction (deterministic).
#pragma unroll
    for (int m = 16; m >= 1; m >>= 1) sum += __shfl_xor(sum, m, 32);
    if (lane == 0) wavePartial[wave] = sum;
    __syncthreads();
    if (tid == 0) {
        float s = 0.0f;
#pragma unroll
        for (int w = 0; w < WAVES; ++w) s += wavePartial[w];
        blockPartial[blockIdx.x] = s;
    }
}

// ---------------------------------------------------------------------------
// Kernel 3: deterministic final reduction of block partials.
// ---------------------------------------------------------------------------
__global__ void final_reduce_kernel(const float* __restrict__ blockPartial,
                                    float* __restrict__ out, int n) {
    if (blockIdx.x == 0 && threadIdx.x == 0) {
        float s = 0.0f;
        for (int i = 0; i < n; ++i) s += blockPartial[i];
        out[0] = 0.5f * s;
    }
}

extern "C" void kernel_launch(void* const* d_in, const int* in_sizes, int n_in,
                              void* d_out, int out_size, void* d_ws, size_t ws_size,
                              hipStream_t stream) {
    const float* src = (const float*)d_in[0];   // [N_SRC, 3] f32
    const float* tar = (const float*)d_in[1];   // [M_TAR, 3] f32

    // Workspace: 256 KB packed sources + 1 KB block partials.
    v4f*   packed   = (v4f*)d_ws;
    float* partials = (float*)((char*)d_ws + (size_t)N_SRC * sizeof(v4f));

    pack_src_kernel<<<N_SRC / 256, 256, 0, stream>>>(src, packed, N_SRC);
    nn_argmin_kernel<<<NBLOCKS, THREADS, 0, stream>>>(src, tar, packed, partials);
    final_reduce_kernel<<<1, 1, 0, stream>>>(partials, (float*)d_out, NBLOCKS);
}